// TaylorActivation_53523882443516
// MI455X (gfx1250) — compile-verified
//
#include <hip/hip_runtime.h>

// Order-8 polynomial (Horner) applied elementwise to 512x65536 fp32.
// HBM-bandwidth-bound: 256 MiB traffic, AI = 2 flop/B, floor ~11.5us @ 23.3TB/s.
// CDNA5 path: deep-pipelined async global->LDS B128 loads (ASYNCcnt, 7 in
// flight per wave = 3.5KB -> aggregate in-flight >> BW*latency product),
// ds_load_b128 -> 8 VALU FMAs/elem -> non-temporal B128 stores.

#define ORDER 8

typedef float v4f __attribute__((ext_vector_type(4)));
typedef int   v4i __attribute__((ext_vector_type(4)));

#if defined(__has_builtin)
#if __has_builtin(__builtin_amdgcn_global_load_async_to_lds_b128) && \
    __has_builtin(__builtin_amdgcn_s_wait_asynccnt)
#define USE_ASYNC_LDS 1
#endif
#endif

__device__ __forceinline__ v4f horner4(v4f xe, const float* c) {
  v4f acc = {c[ORDER], c[ORDER], c[ORDER], c[ORDER]};
#pragma unroll
  for (int k = ORDER - 1; k >= 0; --k) acc = acc * xe + c[k];
  return acc;
}

// ---------------------------------------------------------------------------
// Main kernel: 8-buffer async global->LDS B128 pipeline, 7 loads in flight.
// Each 256-thread block streams CHUNKS tiles of 256 float4 (4 KB LDS each).
// Per-lane LDS slots only -> per-wave ASYNCcnt wait suffices, no barriers.
// ---------------------------------------------------------------------------
#ifdef USE_ASYNC_LDS

#define TILE   256
#define CHUNKS 16   // tiles per block
#define NBUF   8    // LDS buffers (32 KB)
#define DEPTH  7    // outstanding async loads in steady state

// Builtin params are int4 pointers in AS1 (global) / AS3 (LDS).
typedef __attribute__((address_space(1))) v4i as1_v4i;
typedef __attribute__((address_space(3))) v4i as3_v4i;

__global__ __launch_bounds__(256) void taylor_async_kernel(
    const float* __restrict__ x, const float* __restrict__ w,
    float* __restrict__ out) {
  __shared__ v4f tile[NBUF][TILE];

  float c[ORDER + 1];
#pragma unroll
  for (int i = 0; i <= ORDER; ++i) c[i] = w[i];

  const v4f* __restrict__ xv = (const v4f*)x;
  v4f* __restrict__ ov = (v4f*)out;
  const int tid = threadIdx.x;
  const long long base = (long long)blockIdx.x * (TILE * CHUNKS) + tid;

  // Prologue: fill the pipeline with DEPTH async tile loads.
#pragma unroll
  for (int j = 0; j < DEPTH; ++j) {
    __builtin_amdgcn_global_load_async_to_lds_b128(
        (as1_v4i*)(xv + base + (long long)j * TILE),
        (as3_v4i*)&tile[j][tid], 0, 0);
  }

#pragma unroll
  for (int j = 0; j < CHUNKS; ++j) {
    if (j + DEPTH < CHUNKS) {
      // Keep DEPTH loads in flight; in-order completion within a wave means
      // asynccnt<=DEPTH guarantees tile j has landed in LDS.
      __builtin_amdgcn_global_load_async_to_lds_b128(
          (as1_v4i*)(xv + base + (long long)(j + DEPTH) * TILE),
          (as3_v4i*)&tile[(j + DEPTH) % NBUF][tid], 0, 0);
      __builtin_amdgcn_s_wait_asynccnt(DEPTH);
    } else {
      // Tail: drain everything once (loads were issued long ago; after the
      // first drain this is free).
      __builtin_amdgcn_s_wait_asynccnt(0);
    }
    v4f xe = tile[j % NBUF][tid];     // ds_load_b128
    v4f r = horner4(xe, c);           // 8 FMAs/lane-element
    __builtin_nontemporal_store(r, &ov[base + (long long)j * TILE]);
  }
}
#endif  // USE_ASYNC_LDS

// ---------------------------------------------------------------------------
// Fallback / tail kernel: grid-stride float4 NT streaming.
// ---------------------------------------------------------------------------
__global__ __launch_bounds__(256) void taylor_stream_kernel(
    const float* __restrict__ x, const float* __restrict__ w,
    float* __restrict__ out, long long n, long long vec_start) {
  float c[ORDER + 1];
#pragma unroll
  for (int i = 0; i <= ORDER; ++i) c[i] = w[i];

  const long long nvec = n >> 2;
  const v4f* __restrict__ xv = (const v4f*)x;
  v4f* __restrict__ ov = (v4f*)out;
  const long long stride = (long long)gridDim.x * blockDim.x;

  for (long long v = vec_start + (long long)blockIdx.x * blockDim.x + threadIdx.x;
       v < nvec; v += stride) {
    v4f xe = __builtin_nontemporal_load(&xv[v]);
    v4f r = horner4(xe, c);
    __builtin_nontemporal_store(r, &ov[v]);
  }

  // Scalar tail (n % 4 elements), handled by block 0.
  if (blockIdx.x == 0) {
    const int tail = (int)(n & 3);
    if ((int)threadIdx.x < tail) {
      const long long e = (nvec << 2) + threadIdx.x;
      float xe = x[e];
      float acc = c[ORDER];
#pragma unroll
      for (int k = ORDER - 1; k >= 0; --k) acc = acc * xe + c[k];
      out[e] = acc;
    }
  }
}

// ---------------------------------------------------------------------------
extern "C" void kernel_launch(void* const* d_in, const int* in_sizes, int n_in,
                              void* d_out, int out_size, void* d_ws, size_t ws_size,
                              hipStream_t stream) {
  const float* x = (const float*)d_in[0];
  const float* w = (const float*)d_in[1];
  float* out = (float*)d_out;

  const long long n = (long long)in_sizes[0];  // 512*65536 = 33,554,432
  const long long nvec = n >> 2;

  long long vec_done = 0;
#ifdef USE_ASYNC_LDS
  {
    const long long per_block = (long long)TILE * CHUNKS;  // 4096 float4 / block
    const long long ablocks = nvec / per_block;            // 2048 for this shape
    if (ablocks > 0) {
      taylor_async_kernel<<<(unsigned)ablocks, 256, 0, stream>>>(x, w, out);
      vec_done = ablocks * per_block;
    }
  }
#endif
  if (vec_done < nvec || (n & 3)) {
    long long rem = nvec - vec_done;
    long long blocks = (rem + 255) / 256;
    if (blocks < 1) blocks = 1;
    if (blocks > (1LL << 20)) blocks = 1LL << 20;
    taylor_stream_kernel<<<(unsigned)blocks, 256, 0, stream>>>(x, w, out, n, vec_done);
  }
}